// GCN_74955769249951
// MI455X (gfx1250) — compile-verified
//
#include <hip/hip_runtime.h>
#include <hip/hip_bf16.h>

typedef __attribute__((ext_vector_type(2))) float v2f;
typedef __attribute__((ext_vector_type(8))) float v8f;

#define IN_FEATS 64
#define OUT_FEATS 64

// ---------------------------------------------------------------------------
// Kernel 1: zero the aggregation buffer (float4 stores)
// ---------------------------------------------------------------------------
__global__ void gcn_zero_kernel(float4* __restrict__ agg4, int n4) {
    int i = blockIdx.x * blockDim.x + threadIdx.x;
    if (i < n4) {
        agg4[i] = make_float4(0.f, 0.f, 0.f, 0.f);
    }
}

// ---------------------------------------------------------------------------
// Kernel 2: edge scatter  agg[dst] += feat[src]
// 16 lanes per edge, each lane moves one float4 (256B coalesced per edge).
// atomicAdd(f32) lowers to global_atomic_add_f32 (no return -> STOREcnt).
// feat/agg both fit in 192MB L2, so this runs at L2 bandwidth.
// ---------------------------------------------------------------------------
__global__ void gcn_scatter_kernel(const float4* __restrict__ feat4,
                                   const int* __restrict__ src,
                                   const int* __restrict__ dst,
                                   float* __restrict__ agg,
                                   int n_edges) {
    int gid  = blockIdx.x * blockDim.x + threadIdx.x;
    int edge = gid >> 4;        // 16 lanes per edge
    int sub  = gid & 15;        // which float4 of the 64-float feature row
    if (edge >= n_edges) return;

    int s = src[edge];
    int d = dst[edge];
    float4 v = feat4[(size_t)s * 16 + sub];
    float* p = agg + (size_t)d * IN_FEATS + sub * 4;
    atomicAdd(p + 0, v.x);
    atomicAdd(p + 1, v.y);
    atomicAdd(p + 2, v.z);
    atomicAdd(p + 3, v.w);
}

// ---------------------------------------------------------------------------
// Kernel 3: out = relu(agg @ W^T + b) via V_WMMA_F32_16X16X4_F32.
// Block = 256 threads = 8 waves (wave32). Each wave computes a 16-row x 64-col
// output tile with 4 16x16 f32 accumulators; K loop = 16 steps of K=4.
// W (64x64) is staged transposed into LDS: Wt[k][n] so B-fragment reads are
// lane-consecutive (conflict-free).
//
// Fragment layouts (ISA 7.12.2, f32):
//   A 16x4 : lane L (L&15 = row M), VGPR0 = K + (L>=16 ? 2:0), VGPR1 = +1
//   B 4x16 : lane L (L&15 = col N), VGPR0 = K + (L>=16 ? 2:0), VGPR1 = +1
//   C/D    : VGPR r -> row (r + (L>=16 ? 8:0)), col = L&15
// ---------------------------------------------------------------------------
__global__ void gcn_gemm_relu_kernel(const float* __restrict__ agg,
                                     const float* __restrict__ W,   // [64][64] row-major
                                     const float* __restrict__ bias,
                                     float* __restrict__ out,
                                     int n_nodes, int n_tiles) {
    __shared__ float Wt[IN_FEATS * OUT_FEATS];   // Wt[k*64 + n] = W[n*64 + k], 16KB

    int tid = threadIdx.x;

    // Stage W transposed into LDS (coalesced global reads along k).
    for (int idx = tid; idx < IN_FEATS * OUT_FEATS; idx += 256) {
        int n = idx >> 6;      // output feature (row of W)
        int k = idx & 63;      // input feature  (col of W)
        Wt[k * OUT_FEATS + n] = W[idx];
    }
    __syncthreads();

    int wave = tid >> 5;            // 0..7
    int lane = tid & 31;
    int tile = blockIdx.x * 8 + wave;
    if (tile >= n_tiles) return;    // after the barrier: safe

    int lrow  = lane & 15;          // row within tile (A) / col within ntile (B,D)
    int lhalf = (lane >> 4);        // 0: K{0,1}/rows 0-7, 1: K{2,3}/rows 8-15

    int row = tile * 16 + lrow;
    if (row >= n_nodes) row = n_nodes - 1;   // clamp (N is a multiple of 16 anyway)
    const float* arow = agg + (size_t)row * IN_FEATS;

    v8f acc0 = {}, acc1 = {}, acc2 = {}, acc3 = {};

    #pragma unroll
    for (int k0 = 0; k0 < IN_FEATS; k0 += 4) {
        int ka = k0 + lhalf * 2;

        v2f a;
        a.x = arow[ka + 0];
        a.y = arow[ka + 1];

        const float* w0 = &Wt[(ka + 0) * OUT_FEATS + lrow];
        const float* w1 = &Wt[(ka + 1) * OUT_FEATS + lrow];

        v2f b0; b0.x = w0[0];  b0.y = w1[0];
        v2f b1; b1.x = w0[16]; b1.y = w1[16];
        v2f b2; b2.x = w0[32]; b2.y = w1[32];
        v2f b3; b3.x = w0[48]; b3.y = w1[48];

        acc0 = __builtin_amdgcn_wmma_f32_16x16x4_f32(false, a, false, b0, (short)0, acc0, false, false);
        acc1 = __builtin_amdgcn_wmma_f32_16x16x4_f32(false, a, false, b1, (short)0, acc1, false, false);
        acc2 = __builtin_amdgcn_wmma_f32_16x16x4_f32(false, a, false, b2, (short)0, acc2, false, false);
        acc3 = __builtin_amdgcn_wmma_f32_16x16x4_f32(false, a, false, b3, (short)0, acc3, false, false);
    }

    // Bias + ReLU + store. D VGPR r: row = tile*16 + r + 8*lhalf, col = ntile*16 + lrow.
    int rbase = tile * 16 + 8 * lhalf;
    #pragma unroll
    for (int j = 0; j < 4; ++j) {
        int   n  = j * 16 + lrow;
        float bv = bias[n];
        const v8f* accp = (j == 0) ? &acc0 : (j == 1) ? &acc1 : (j == 2) ? &acc2 : &acc3;
        v8f accv = *accp;
        #pragma unroll
        for (int r = 0; r < 8; ++r) {
            int rr = rbase + r;
            if (rr < n_nodes) {
                float v = accv[r] + bv;
                out[(size_t)rr * OUT_FEATS + n] = v > 0.f ? v : 0.f;
            }
        }
    }
}

// ---------------------------------------------------------------------------
// Launcher
// inputs: feat [N*64] f32, W [64*64] f32, b [64] f32, src [E] i32, dst [E] i32
// d_ws: agg buffer, N*64 f32 (25.6 MB)
// ---------------------------------------------------------------------------
extern "C" void kernel_launch(void* const* d_in, const int* in_sizes, int n_in,
                              void* d_out, int out_size, void* d_ws, size_t ws_size,
                              hipStream_t stream) {
    const float* feat = (const float*)d_in[0];
    const float* W    = (const float*)d_in[1];
    const float* bias = (const float*)d_in[2];
    const int*   src  = (const int*)d_in[3];
    const int*   dst  = (const int*)d_in[4];
    float*       out  = (float*)d_out;
    float*       agg  = (float*)d_ws;

    int n_nodes = in_sizes[0] / IN_FEATS;
    int n_edges = in_sizes[3];

    // 1) zero agg
    {
        int n4 = n_nodes * (IN_FEATS / 4);
        int blocks = (n4 + 255) / 256;
        gcn_zero_kernel<<<blocks, 256, 0, stream>>>((float4*)agg, n4);
    }

    // 2) scatter-add
    {
        long long threads = (long long)n_edges * 16;
        int blocks = (int)((threads + 255) / 256);
        gcn_scatter_kernel<<<blocks, 256, 0, stream>>>(
            (const float4*)feat, src, dst, agg, n_edges);
    }

    // 3) WMMA GEMM + bias + relu
    {
        int n_tiles = (n_nodes + 15) / 16;         // 6250
        int blocks  = (n_tiles + 7) / 8;           // 8 waves per block
        gcn_gemm_relu_kernel<<<blocks, 256, 0, stream>>>(
            agg, W, bias, out, n_nodes, n_tiles);
    }
}